// ROIPool_61151744360593
// MI455X (gfx1250) — compile-verified
//
#include <hip/hip_runtime.h>

// ROI rotated-grid pooling, collapsed: the align-corners 16->4 resize has
// exactly-integer source coords (15/3 = 5.0), so output only depends on grid
// indices {0,5,10,15}^3 -> 64 points per ROI instead of 4096.
//
// Gather-bound kernel; the per-ROI affine transform of the 64 grid points is
// done with V_WMMA_F32_16X16X4_F32 (A = [R|center] 16x4, B = homogeneous
// points 4x16, D rows 0..2 = p) when the builtin is available on the device
// pass; otherwise a per-lane FMA fallback (identical math).

typedef __attribute__((ext_vector_type(2))) float v2f;
typedef __attribute__((ext_vector_type(8))) float v8f;

#if defined(__HIP_DEVICE_COMPILE__) && __has_builtin(__builtin_amdgcn_wmma_f32_16x16x4_f32)
#define HAVE_WMMA_F32X4 1
#else
#define HAVE_WMMA_F32X4 0
#endif

#define NCH   64
#define DIMX  128
#define DIMY  128
#define DIMZ  128
#define PLANE (DIMX * DIMY * DIMZ)   // elements per channel (2097152)
#define HALFG 7.5f                   // (G-1)/2

__global__ __launch_bounds__(256) void ROIPool_61151744360593_kernel(
    const float* __restrict__ feature,   // (C, X, Y, Z)
    const float* __restrict__ rois,      // (N, 8): level, cx, cy, cz, sx, sy, sz, theta
    float* __restrict__ out)             // (N, C, 4, 4, 4)
{
  const int n    = blockIdx.x;
  const int cgrp = blockIdx.y;               // channel quarter: [16*cgrp, 16*cgrp+16)
  const int t    = (int)threadIdx.x;
  const int lane = t & 31;
  const int M    = lane & 15;                // row (A) or column (B/D) index
  const bool hi  = lane >= 16;
  const int sidx = t & 63;                   // spatial point 0..63 (z fastest)
  const int cb   = (t >> 6) + (cgrp << 4);   // first channel for this thread

  // ---- ROI parameters (uniform per block) ----
  const float cx = rois[n * 8 + 1];
  const float cy = rois[n * 8 + 2];
  const float cz = rois[n * 8 + 3];
  const float th = rois[n * 8 + 7];
  const float ct = cosf(th);
  const float st = sinf(th);

  float px, py, pz;

#if HAVE_WMMA_F32X4
  // ---- A matrix (16x4 f32): rows 0..2 = [R | center], rest zero ----
  // 32-bit A layout: lanes 0-15 hold cols {0,1}; lanes 16-31 hold cols {2,3}.
  const float a0 = (M == 0) ? (hi ? 0.f : ct)
                 : (M == 1) ? (hi ? 0.f : st)
                 : (M == 2) ? (hi ? 1.f : 0.f) : 0.f;
  const float a1 = (M == 0) ? (hi ? cx : -st)
                 : (M == 1) ? (hi ? cy :  ct)
                 : (M == 2) ? (hi ? cz : 0.f) : 0.f;
  v2f A; A.x = a0; A.y = a1;

  // ---- Two WMMAs per wave: 16 homogeneous points each ----
  // B (4x16 f32): column = lane&15; lanes 0-15 hold rows {0,1} = (ax, ay),
  // lanes 16-31 hold rows {2,3} = (az, 1).
  const int s0 = ((t >> 5) & 1) << 5;        // first point of this wave (0 or 32)
  v8f d0, d1;
  {
    int ps = s0 + M;                         // batch 0: points s0 .. s0+15
    float axp = 5.f * (float)(ps >> 4)        - HALFG;
    float ayp = 5.f * (float)((ps >> 2) & 3)  - HALFG;
    float azp = 5.f * (float)(ps & 3)         - HALFG;
    v2f B; B.x = hi ? azp : axp; B.y = hi ? 1.f : ayp;
    v8f c = {};
    d0 = __builtin_amdgcn_wmma_f32_16x16x4_f32(false, A, false, B,
                                               (short)0, c, false, false);
  }
  {
    int ps = s0 + 16 + M;                    // batch 1: points s0+16 .. s0+31
    float axp = 5.f * (float)(ps >> 4)        - HALFG;
    float ayp = 5.f * (float)((ps >> 2) & 3)  - HALFG;
    float azp = 5.f * (float)(ps & 3)         - HALFG;
    v2f B; B.x = hi ? azp : axp; B.y = hi ? 1.f : ayp;
    v8f c = {};
    d1 = __builtin_amdgcn_wmma_f32_16x16x4_f32(false, A, false, B,
                                               (short)0, c, false, false);
  }

  // ---- Redistribute: this lane's point is s = s0 + lane -> column (lane&15)
  // of batch (lane>=16). D: lane j (0..15), VGPR r holds row r, col j.
  // Shuffle both batches unconditionally (no EXEC-masked source lanes).
  float x0 = __shfl(d0[0], M, 32), x1 = __shfl(d1[0], M, 32);
  float y0 = __shfl(d0[1], M, 32), y1 = __shfl(d1[1], M, 32);
  float z0 = __shfl(d0[2], M, 32), z1 = __shfl(d1[2], M, 32);
  px = hi ? x1 : x0;
  py = hi ? y1 : y0;
  pz = hi ? z1 : z0;
#else
  // ---- Per-lane FMA fallback: same affine transform ----
  {
    float axp = 5.f * (float)(sidx >> 4)       - HALFG;
    float ayp = 5.f * (float)((sidx >> 2) & 3) - HALFG;
    float azp = 5.f * (float)(sidx & 3)        - HALFG;
    px = ct * axp - st * ayp + cx;
    py = st * axp + ct * ayp + cy;
    pz = azp + cz;
  }
#endif

  // ---- In-bounds mask folded into the final scale ----
  const bool inb = (px >= 0.f) && (px <= (float)(DIMX - 1)) &&
                   (py >= 0.f) && (py <= (float)(DIMY - 1)) &&
                   (pz >= 0.f) && (pz <= (float)(DIMZ - 1));
  const float scale = inb ? 0.125f : 0.f;

  // ---- 8 corners: weights from UNclipped floor/ceil, offsets from clipped ----
  const float fx = floorf(px), gx = ceilf(px);
  const float fy = floorf(py), gy = ceilf(py);
  const float fz = floorf(pz), gz = ceilf(pz);
  const float dxf = fabsf(px - fx), dxc = fabsf(px - gx);
  const float dyf = fabsf(py - fy), dyc = fabsf(py - gy);
  const float dzf = fabsf(pz - fz), dzc = fabsf(pz - gz);
  const int oxf = ((int)fminf(fmaxf(fx, 0.f), (float)(DIMX - 1))) << 14;
  const int oxc = ((int)fminf(fmaxf(gx, 0.f), (float)(DIMX - 1))) << 14;
  const int oyf = ((int)fminf(fmaxf(fy, 0.f), (float)(DIMY - 1))) << 7;
  const int oyc = ((int)fminf(fmaxf(gy, 0.f), (float)(DIMY - 1))) << 7;
  const int ozf =  (int)fminf(fmaxf(fz, 0.f), (float)(DIMZ - 1));
  const int ozc =  (int)fminf(fmaxf(gz, 0.f), (float)(DIMZ - 1));

  int   off[8];
  float w[8];
  off[0] = oxf + oyf + ozf;  w[0] = 1.f - dxf * dyf * dzf;
  off[1] = oxf + oyf + ozc;  w[1] = 1.f - dxf * dyf * dzc;
  off[2] = oxf + oyc + ozf;  w[2] = 1.f - dxf * dyc * dzf;
  off[3] = oxf + oyc + ozc;  w[3] = 1.f - dxf * dyc * dzc;
  off[4] = oxc + oyf + ozf;  w[4] = 1.f - dxc * dyf * dzf;
  off[5] = oxc + oyf + ozc;  w[5] = 1.f - dxc * dyf * dzc;
  off[6] = oxc + oyc + ozf;  w[6] = 1.f - dxc * dyc * dzf;
  off[7] = oxc + oyc + ozc;  w[7] = 1.f - dxc * dyc * dzc;

  // ---- Gather loop over this thread's 4 channels ----
  const size_t obase = ((size_t)n << 12) + (size_t)sidx;   // n*C*64 + s
  #pragma unroll
  for (int k = 0; k < 4; ++k) {
    const int cc = cb + (k << 2);
    const float* __restrict__ fp = feature + (size_t)cc * (size_t)PLANE;
    float acc = 0.f;
    #pragma unroll
    for (int u = 0; u < 8; ++u)
      acc += fp[off[u]] * w[u];
    out[obase + ((size_t)cc << 6)] = acc * scale;          // coalesced in sidx
  }
}

extern "C" void kernel_launch(void* const* d_in, const int* in_sizes, int n_in,
                              void* d_out, int out_size, void* d_ws, size_t ws_size,
                              hipStream_t stream) {
  const float* feature = (const float*)d_in[0];  // (64, 128, 128, 128) f32
  const float* rois    = (const float*)d_in[1];  // (N, 8) f32
  float*       out     = (float*)d_out;          // (N, 64, 4, 4, 4) f32

  const int N = in_sizes[1] / 8;                 // 256
  dim3 grid((unsigned)N, 4, 1);                  // (ROI, channel-quarter)
  ROIPool_61151744360593_kernel<<<grid, 256, 0, stream>>>(feature, rois, out);
}